// GraphSAGE_90984587198486
// MI455X (gfx1250) — compile-verified
//
#include <hip/hip_runtime.h>
#include <hip/hip_bf16.h>

typedef __attribute__((ext_vector_type(2))) float v2f;
typedef __attribute__((ext_vector_type(8))) float v8f;

#define DIMF 128

__global__ void sage_zero4(float4* __restrict__ p, long long n4) {
    long long i = (long long)blockIdx.x * blockDim.x + threadIdx.x;
    long long stride = (long long)gridDim.x * blockDim.x;
    float4 z = make_float4(0.f, 0.f, 0.f, 0.f);
    for (; i < n4; i += stride) p[i] = z;
}

__global__ void sage_scatter(const float* __restrict__ feat,
                             const int*   __restrict__ src,
                             const int*   __restrict__ dst,
                             float*       __restrict__ msg,
                             float*       __restrict__ deg,
                             int nE) {
    long long tid = (long long)blockIdx.x * blockDim.x + threadIdx.x;
    int e    = (int)(tid >> 5);
    int lane = (int)(tid & 31);
    if (e >= nE) return;
    int s = src[e];
    int d = dst[e];
    float4 v = reinterpret_cast<const float4*>(feat + (long long)s * DIMF)[lane];
    float* base = msg + (long long)d * DIMF + lane * 4;
    atomicAdd(base + 0, v.x);
    atomicAdd(base + 1, v.y);
    atomicAdd(base + 2, v.z);
    atomicAdd(base + 3, v.w);
    if (deg != nullptr && lane == 0) atomicAdd(deg + d, 1.0f);
}

template <bool RELU>
__global__ void sage_gemm(const float* __restrict__ hin,
                          const float* __restrict__ msg,
                          const float* __restrict__ deg,
                          const float* __restrict__ Wself,
                          const float* __restrict__ Wneigh,
                          const float* __restrict__ bias,
                          float*       __restrict__ out,
                          int nN) {
    __shared__ float As[16 * 260];

    const int rowbase = blockIdx.x * 16;
    const int t = threadIdx.x;

    {
        int row  = t >> 4;
        int colb = (t & 15) * 16;
        int r = rowbase + row;
        if (r < nN) {
            float rdeg = 1.0f / fmaxf(deg[r], 1.0f);
            const float4* xr = reinterpret_cast<const float4*>(hin + (long long)r * DIMF);
            const float4* mr = reinterpret_cast<const float4*>(msg + (long long)r * DIMF);
#pragma unroll
            for (int i = 0; i < 4; ++i) {
                int col = colb + i * 4;
                float4 v;
                if (col < 128) {
                    v = xr[col >> 2];
                } else {
                    float4 m = mr[(col - 128) >> 2];
                    v = make_float4(m.x * rdeg, m.y * rdeg, m.z * rdeg, m.w * rdeg);
                }
                *reinterpret_cast<float4*>(&As[row * 260 + col]) = v;
            }
        } else {
            float4 z = make_float4(0.f, 0.f, 0.f, 0.f);
#pragma unroll
            for (int i = 0; i < 4; ++i)
                *reinterpret_cast<float4*>(&As[row * 260 + colb + i * 4]) = z;
        }
    }
    __syncthreads();

    const int wave = t >> 5;
    const int lane = t & 31;
    const int nl   = lane & 15;
    const int half = lane >> 4;
    const int ncol = wave * 16 + nl;

    v8f c;
    {
        float bv = bias[ncol];
#pragma unroll
        for (int j = 0; j < 8; ++j) c[j] = bv;
    }

#pragma unroll 4
    for (int k = 0; k < 128; k += 4) {
        int kh = k + half * 2;
        v2f a = { As[nl * 260 + kh], As[nl * 260 + kh + 1] };
        v2f b = { Wself[kh * DIMF + ncol], Wself[(kh + 1) * DIMF + ncol] };
        c = __builtin_amdgcn_wmma_f32_16x16x4_f32(false, a, false, b,
                                                  (short)0, c, false, false);
    }
#pragma unroll 4
    for (int k = 0; k < 128; k += 4) {
        int kh = k + half * 2;
        v2f a = { As[nl * 260 + 128 + kh], As[nl * 260 + 128 + kh + 1] };
        v2f b = { Wneigh[kh * DIMF + ncol], Wneigh[(kh + 1) * DIMF + ncol] };
        c = __builtin_amdgcn_wmma_f32_16x16x4_f32(false, a, false, b,
                                                  (short)0, c, false, false);
    }

#pragma unroll
    for (int j = 0; j < 8; ++j) {
        int row = rowbase + j + half * 8;
        if (row < nN) {
            float v = c[j];
            if (RELU) v = fmaxf(v, 0.0f);
            out[(long long)row * DIMF + ncol] = v;
        }
    }
}

extern "C" void kernel_launch(void* const* d_in, const int* in_sizes, int n_in,
                              void* d_out, int out_size, void* d_ws, size_t ws_size,
                              hipStream_t stream) {
    const float* x   = (const float*)d_in[0];
    const int*   src = (const int*)d_in[1];
    const int*   dst = (const int*)d_in[2];
    const float* Ws1 = (const float*)d_in[3];
    const float* Wn1 = (const float*)d_in[4];
    const float* b1  = (const float*)d_in[5];
    const float* Ws2 = (const float*)d_in[6];
    const float* Wn2 = (const float*)d_in[7];
    const float* b2  = (const float*)d_in[8];
    float* out = (float*)d_out;

    const int nN = in_sizes[0] / DIMF;   // 50000
    const int nE = in_sizes[1];          // 800000

    const long long NPAD = ((long long)nN + 63) & ~63LL;
    float* deg = (float*)d_ws;
    float* msg = deg + NPAD;
    float* h   = msg + (long long)nN * DIMF;

    const int ZB = 256;
    long long nz0 = (NPAD + (long long)nN * DIMF) / 4;
    long long nz1 = ((long long)nN * DIMF) / 4;

    int zgrid0 = (int)((nz0 + ZB - 1) / ZB); if (zgrid0 > 16384) zgrid0 = 16384;
    int zgrid1 = (int)((nz1 + ZB - 1) / ZB); if (zgrid1 > 16384) zgrid1 = 16384;

    long long sthreads = (long long)nE * 32;
    int sgrid = (int)((sthreads + 255) / 256);
    int ggrid = (nN + 15) / 16;

    // Layer 1
    sage_zero4<<<zgrid0, ZB, 0, stream>>>((float4*)d_ws, nz0);
    sage_scatter<<<sgrid, 256, 0, stream>>>(x, src, dst, msg, deg, nE);
    sage_gemm<true><<<ggrid, 256, 0, stream>>>(x, msg, deg, Ws1, Wn1, b1, h, nN);

    // Layer 2 (deg reused from layer 1)
    sage_zero4<<<zgrid1, ZB, 0, stream>>>((float4*)msg, nz1);
    sage_scatter<<<sgrid, 256, 0, stream>>>(h, src, dst, msg, nullptr, nE);
    sage_gemm<false><<<ggrid, 256, 0, stream>>>(h, msg, deg, Ws2, Wn2, b2, out, nN);
}